// MaskAdaptiveEmbedding_56908316672214
// MI455X (gfx1250) — compile-verified
//
#include <hip/hip_runtime.h>

// ---------------------------------------------------------------------------
// Adaptive embedding for MI455X (gfx1250, wave32).
//   out[t, :] = scale * ( cluster0 ? emb0[tok]             (copy)
//                                  : emb_i[tok-l] @ proj_i (WMMA f32 16x16x4) )
// Memory-bound (~73 MiB traffic -> ~3us @ 23.3 TB/s); WMMA does the projection
// in full fp32 (reference dtype) while the memory system streams gathers and
// the 64 MiB output. Store path is branch-free for full 16-token tiles.
// ---------------------------------------------------------------------------

#define NTOK  16384      // B*S = 8*2048
#define DPROJ 1024

typedef __attribute__((ext_vector_type(2))) float v2f;
typedef __attribute__((ext_vector_type(8))) float v8f;

// --- pass 0: zero the per-cluster counters in workspace ---------------------
__global__ void ae_zero_kernel(int* counts) {
    if (threadIdx.x < 4) counts[threadIdx.x] = 0;
}

// --- pass 1: partition tokens into 4 compacted per-cluster lists ------------
__global__ __launch_bounds__(256)
void ae_partition_kernel(const int* __restrict__ tokens,
                         int* __restrict__ counts,
                         int* __restrict__ lists) {
    int t = blockIdx.x * 256 + threadIdx.x;
    if (t >= NTOK) return;
    int tok = tokens[t];
    int c = (tok < 20000) ? 0 : (tok < 40000) ? 1 : (tok < 200000) ? 2 : 3;
    int pos = atomicAdd(&counts[c], 1);
    lists[c * NTOK + pos] = t;            // order irrelevant: each token owns its out row
}

// --- cluster 0: de == 1024, pure scaled copy (b128 vector path) -------------
__global__ __launch_bounds__(256)
void ae_cluster0_kernel(const int* __restrict__ tokens,
                        const int* __restrict__ counts,
                        const int* __restrict__ lists,
                        const float* __restrict__ emb0,
                        float* __restrict__ out) {
    int b = blockIdx.x;
    if (b >= counts[0]) return;           // uniform early-exit
    int t   = lists[b];                   // cluster-0 list at offset 0
    int row = tokens[t];                  // cutoff is 0
    const float4* src = (const float4*)emb0 + (size_t)row * (DPROJ / 4);
    float4*       dst = (float4*)out     + (size_t)t   * (DPROJ / 4);
    float4 v = src[threadIdx.x];          // global_load_b128
    v.x *= 32.0f; v.y *= 32.0f; v.z *= 32.0f; v.w *= 32.0f;
    dst[threadIdx.x] = v;                 // global_store_b128
}

// --- clusters 1..3: [16 tokens x DE] @ [DE x 1024] via V_WMMA_F32_16X16X4_F32
// Block = 256 threads = 8 waves. Each wave owns a 128-column strip (8 N-tiles
// of 16). A-tile staged in LDS with padded stride (DE+2) so the per-K-step
// ds_load_b64 (lanes 0-15: A[m][k..k+1], lanes 16-31: A[m][k+2..k+3]) is
// bank-conflict free. B comes straight from proj (tiny, L2-resident):
//   B VGPR0 = proj[k + 2*half][N0 + lane%16], VGPR1 = next K row.
// Stores: full tiles (the common case) take an unconditional, branch-free
// path; only the single ragged tail tile per cluster uses per-row predication.
template <int DE>
__global__ __launch_bounds__(256)
void ae_gemm_kernel(const int* __restrict__ tokens,
                    const int* __restrict__ counts,
                    const int* __restrict__ lists,
                    int c, int cutoff,
                    const float* __restrict__ emb,
                    const float* __restrict__ proj,
                    float* __restrict__ out) {
    constexpr int LDA = DE + 2;           // padded LDS row stride (even -> 8B aligned)
    __shared__ float Alds[16 * LDA];
    __shared__ int   toks[16];

    int cnt  = counts[c];
    int tile = blockIdx.x;                // 16 tokens per tile
    if (tile * 16 >= cnt) return;         // uniform per block -> EXEC full at WMMA
    bool full = (tile * 16 + 16 <= cnt);  // block-uniform: all 16 rows valid?

    int tid = threadIdx.x;
    if (tid < 16) {
        int j = tile * 16 + tid;
        toks[tid] = (j < cnt) ? lists[c * NTOK + j] : -1;
    }
    __syncthreads();

    // cooperative gather of the 16 embedding rows into LDS (zero-pad ragged tile)
    for (int idx = tid; idx < 16 * DE; idx += 256) {
        int m = idx / DE, k = idx % DE;
        int t = toks[m];
        float v = 0.0f;
        if (t >= 0) {
            int row = tokens[t] - cutoff;
            v = emb[(size_t)row * DE + k];
        }
        Alds[m * LDA + k] = v;
    }
    __syncthreads();

    int lane = tid & 31;
    int wave = tid >> 5;
    int half = lane >> 4;                 // 0: K rows {k,k+1}, 1: K rows {k+2,k+3}
    int l16  = lane & 15;
    const float* arow = &Alds[l16 * LDA];

    // hoist this lane's 8 output-row ids out of the nt loop (8 ds_loads total)
    int tgs[8];
#pragma unroll
    for (int r = 0; r < 8; ++r) tgs[r] = toks[r + half * 8];

    for (int nt = 0; nt < 8; ++nt) {
        int N0 = wave * 128 + nt * 16;
        v8f acc = {};
        for (int k = 0; k < DE; k += 4) {
            int ka = k + half * 2;
            v2f a = *(const v2f*)(arow + ka);                      // ds_load_b64
            v2f b;
            b.x = proj[(size_t)ka * DPROJ + N0 + l16];             // L2-hit loads
            b.y = proj[(size_t)(ka + 1) * DPROJ + N0 + l16];
            acc = __builtin_amdgcn_wmma_f32_16x16x4_f32(
                      /*neg_a=*/false, a, /*neg_b=*/false, b,
                      /*c_mod=*/(short)0, acc,
                      /*reuse_a=*/false, /*reuse_b=*/false);
        }
        // C/D layout: VGPR r -> M = r + 8*half, N = N0 + lane%16
        if (full) {
#pragma unroll
            for (int r = 0; r < 8; ++r)
                out[(size_t)tgs[r] * DPROJ + N0 + l16] = acc[r] * 32.0f;
        } else {
#pragma unroll
            for (int r = 0; r < 8; ++r)
                if (tgs[r] >= 0)
                    out[(size_t)tgs[r] * DPROJ + N0 + l16] = acc[r] * 32.0f;
        }
    }
}

// ---------------------------------------------------------------------------
extern "C" void kernel_launch(void* const* d_in, const int* in_sizes, int n_in,
                              void* d_out, int out_size, void* d_ws, size_t ws_size,
                              hipStream_t stream) {
    // setup_inputs() order: tokens, emb0, emb1, proj1, emb2, proj2, emb3, proj3
    const int*   tokens = (const int*)  d_in[0];
    const float* emb0   = (const float*)d_in[1];
    const float* emb1   = (const float*)d_in[2];
    const float* proj1  = (const float*)d_in[3];
    const float* emb2   = (const float*)d_in[4];
    const float* proj2  = (const float*)d_in[5];
    const float* emb3   = (const float*)d_in[6];
    const float* proj3  = (const float*)d_in[7];
    float* out = (float*)d_out;

    // workspace: counts[4] (padded to 64 ints) + 4 compacted token lists
    int* counts = (int*)d_ws;
    int* lists  = counts + 64;            // 4 * NTOK ints  (~256 KB total)

    ae_zero_kernel<<<1, 32, 0, stream>>>(counts);
    ae_partition_kernel<<<NTOK / 256, 256, 0, stream>>>(tokens, counts, lists);

    // cluster 0: worst-case grid, device-side early exit on count
    ae_cluster0_kernel<<<NTOK, 256, 0, stream>>>(tokens, counts, lists, emb0, out);

    // clusters 1..3: 16-token tiles, worst-case NTOK/16 tiles each
    ae_gemm_kernel<256><<<NTOK / 16, 256, 0, stream>>>(tokens, counts, lists,
                                                       1, 20000,  emb1, proj1, out);
    ae_gemm_kernel< 64><<<NTOK / 16, 256, 0, stream>>>(tokens, counts, lists,
                                                       2, 40000,  emb2, proj2, out);
    ae_gemm_kernel< 16><<<NTOK / 16, 256, 0, stream>>>(tokens, counts, lists,
                                                       3, 200000, emb3, proj3, out);
}